// InstructionTracePositionEmbedding_30142080483470
// MI455X (gfx1250) — compile-verified
//
#include <hip/hip_runtime.h>

typedef __attribute__((ext_vector_type(2))) float v2f;
typedef __attribute__((ext_vector_type(8))) float v8f;

#define BATCH 8
#define SEQ   8192
#define HID   768
#define ROWPAD 772          // 768 + 4 pad -> stride mod 64 banks = 4, conflict-free
#define NEXT_TOKEN_ID 2

// ---------------------------------------------------------------------------
// Kernel 1: per-batch segment scan (cumsum of starts, cummax of start pos).
// One wave32 per batch row; shuffle scan with serial carry over 256 chunks.
// ---------------------------------------------------------------------------
__global__ __launch_bounds__(32) void itpe_scan_kernel(
    const int* __restrict__ ids,
    int* __restrict__ instrIdx,
    int* __restrict__ argIdx) {
  const int b = blockIdx.x;
  const int lane = threadIdx.x;
  const int* row = ids + (size_t)b * SEQ;
  int carryCum = 0;   // instruction counter carry
  int carryLast = 0;  // last segment start position carry
  for (int c = 0; c < SEQ / 32; ++c) {
    const int pos = c * 32 + lane;
    const int prev = (pos == 0) ? -1 : row[pos - 1];
    const int start = (pos > 0 && prev == NEXT_TOKEN_ID) ? 1 : 0;
    int s = start;                 // inclusive prefix sum of starts
    int v = start ? pos : 0;       // inclusive prefix max of start positions
#pragma unroll
    for (int off = 1; off < 32; off <<= 1) {
      const int ns = __shfl_up(s, off, 32);
      const int nv = __shfl_up(v, off, 32);
      if (lane >= off) { s += ns; v = (v > nv) ? v : nv; }
    }
    const int instr = carryCum + s;
    const int last  = (carryLast > v) ? carryLast : v;
    instrIdx[(size_t)b * SEQ + pos] = instr;
    argIdx[(size_t)b * SEQ + pos]   = pos - last;
    carryCum  = __shfl(instr, 31, 32);
    carryLast = __shfl(last, 31, 32);
  }
}

// ---------------------------------------------------------------------------
// Kernel 2: fused triple-gather + sum + LayerNorm for a 16-token tile.
// 256 threads (8 waves). LN row reductions via V_WMMA_F32_16X16X4_F32:
//   sum   : D += A(16x4) x ones(4x16)        -> every column = row sum
//   sumsq : D += A(16x4) x A^T(4x16)         -> Gram; diag = row sum-of-squares
// (A and B VGPR layouts are mutual transposes, so passing the same register
//  pair for SRC0/SRC1 yields A·A^T.)
// ---------------------------------------------------------------------------
__global__ __launch_bounds__(256) void itpe_embed_ln_kernel(
    const int* __restrict__ ids,
    const float* __restrict__ tokEmb,
    const float* __restrict__ insEmb,
    const float* __restrict__ argEmb,
    const float* __restrict__ normW,
    const float* __restrict__ normB,
    const int* __restrict__ instrIdx,
    const int* __restrict__ argIdx,
    float* __restrict__ out) {
  __shared__ float smem[16 * ROWPAD];
  __shared__ float s_sum[16];
  __shared__ float s_sq[16];

  const int tid  = threadIdx.x;
  const int lane = tid & 31;
  const int wave = tid >> 5;

  if (tid < 16) { s_sum[tid] = 0.0f; s_sq[tid] = 0.0f; }

  // ---- Phase A: coalesced triple gather + sum -> LDS tile (16 x 768) ----
  const int m   = tid >> 4;     // token within tile (0..15)
  const int l16 = tid & 15;     // lane within token group
  const size_t g = (size_t)blockIdx.x * 16 + m;   // flat token index in [0, B*S)
  const float* tp = tokEmb + (size_t)ids[g]      * HID;
  const float* ip = insEmb + (size_t)instrIdx[g] * HID;
  const float* ap = argEmb + (size_t)argIdx[g]   * HID;
  float* dst = &smem[m * ROWPAD];
#pragma unroll
  for (int j = 0; j < 12; ++j) {
    const int col = l16 * 4 + j * 64;
    const float4 a = *reinterpret_cast<const float4*>(tp + col);
    const float4 b = *reinterpret_cast<const float4*>(ip + col);
    const float4 c = *reinterpret_cast<const float4*>(ap + col);
    float4 r;
    r.x = a.x + b.x + c.x;
    r.y = a.y + b.y + c.y;
    r.z = a.z + b.z + c.z;
    r.w = a.w + b.w + c.w;
    *reinterpret_cast<float4*>(dst + col) = r;
  }
  __syncthreads();

  // ---- Phase B: WMMA row reductions (EXEC all-1s through the whole loop) ----
  {
    const int r    = lane & 15;         // A-matrix row held by this lane
    const int koff = (lane >> 4) * 2;   // half-wave K split (K+0/1 vs K+2/3)
    v8f dsum = {};
    v8f dsq  = {};
    const v2f ones = {1.0f, 1.0f};
    const float* arow = &smem[r * ROWPAD + koff];
    for (int c = 0; c < 24; ++c) {
      const int k0 = (wave * 24 + c) * 4;
      const v2f a = *reinterpret_cast<const v2f*>(arow + k0);
      dsum = __builtin_amdgcn_wmma_f32_16x16x4_f32(
          false, a, false, ones, (short)0, dsum, false, false);
      dsq = __builtin_amdgcn_wmma_f32_16x16x4_f32(
          false, a, false, a, (short)0, dsq, false, false);
    }
    // D layout: VGPR g holds (M=g, N=lane) for lanes 0-15 and
    //           (M=8+g, N=lane-16) for lanes 16-31.
#pragma unroll
    for (int gg = 0; gg < 8; ++gg) {
      if (lane == gg)      atomicAdd(&s_sq[gg],      dsq[gg]);   // diag row gg
      if (lane == 24 + gg) atomicAdd(&s_sq[8 + gg],  dsq[gg]);   // diag row 8+gg
      if (lane == 0)       atomicAdd(&s_sum[gg],     dsum[gg]);  // rowsum gg
      if (lane == 16)      atomicAdd(&s_sum[8 + gg], dsum[gg]);  // rowsum 8+gg
    }
  }
  __syncthreads();

  // ---- Phase C: normalize + affine, coalesced B128 stores ----
  const float inv = 1.0f / (float)HID;
  const float mu  = s_sum[m] * inv;
  const float var = s_sq[m] * inv - mu * mu;    // biased variance (ref semantics)
  const float rstd = rsqrtf(var + 1e-12f);
  float* op = out + g * HID;
#pragma unroll
  for (int j = 0; j < 12; ++j) {
    const int col = l16 * 4 + j * 64;
    const float4 v  = *reinterpret_cast<const float4*>(dst + col);
    const float4 wv = *reinterpret_cast<const float4*>(normW + col);
    const float4 bv = *reinterpret_cast<const float4*>(normB + col);
    float4 r;
    r.x = (v.x - mu) * rstd * wv.x + bv.x;
    r.y = (v.y - mu) * rstd * wv.y + bv.y;
    r.z = (v.z - mu) * rstd * wv.z + bv.z;
    r.w = (v.w - mu) * rstd * wv.w + bv.w;
    *reinterpret_cast<float4*>(op + col) = r;
  }
}

extern "C" void kernel_launch(void* const* d_in, const int* in_sizes, int n_in,
                              void* d_out, int out_size, void* d_ws, size_t ws_size,
                              hipStream_t stream) {
  const int*   ids = (const int*)d_in[0];
  const float* tok = (const float*)d_in[1];
  const float* ins = (const float*)d_in[2];
  const float* arg = (const float*)d_in[3];
  const float* w   = (const float*)d_in[4];
  const float* b   = (const float*)d_in[5];

  int* instrIdx = (int*)d_ws;                  // B*S ints
  int* argIdx   = instrIdx + BATCH * SEQ;      // B*S ints (512 KB total)

  itpe_scan_kernel<<<BATCH, 32, 0, stream>>>(ids, instrIdx, argIdx);
  itpe_embed_ln_kernel<<<(BATCH * SEQ) / 16, 256, 0, stream>>>(
      ids, tok, ins, arg, w, b, instrIdx, argIdx, (float*)d_out);
}